// Softpool_74869869904665
// MI455X (gfx1250) — compile-verified
//
#include <hip/hip_runtime.h>

#define B_ 16
#define D_ 256
#define N_ 8192
#define R_ 16
#define K_ 2048   // N_/4 (SP_RATIO = 4)

typedef __attribute__((ext_vector_type(2))) float v2f;
typedef __attribute__((ext_vector_type(4))) float v4f;
typedef __attribute__((ext_vector_type(8))) float v8f;
typedef __attribute__((ext_vector_type(4))) int   v4i;

// ---------------------------------------------------------------------------
// Kernel 1: val[b,r,n] = sum_d W[r,d]*x[b,d,n] + bias[r]  via V_WMMA_F32_16X16X4_F32
//           + argmax over r (tie -> lowest r) -> idx_sort (stored as float)
// One wave (32 threads) per (b, 16-point tile). 8192 waves total.
// ---------------------------------------------------------------------------
__global__ __launch_bounds__(32) void softpool_gemm_argmax(
    const float* __restrict__ x, const float* __restrict__ W,
    const float* __restrict__ bias, float* __restrict__ val_ws,
    float* __restrict__ idx_sort_out)
{
    const int tile = blockIdx.x;            // 0 .. B_*N_/16 - 1
    const int b    = tile / (N_ / 16);
    const int n0   = (tile % (N_ / 16)) * 16;
    const int lane = threadIdx.x;           // wave32
    const int m    = lane & 15;             // A row (r), B/C column (n)
    const int hi   = lane >> 4;             // half-wave selector
    const int koff = hi * 2;                // K sub-offset within 16x16x4 step

    const float* xb = x + (size_t)b * D_ * N_;

    // Accumulator init with bias: C VGPR v holds row r = v + 8*hi.
    v8f c;
#pragma unroll
    for (int v = 0; v < 8; ++v) c[v] = bias[v + 8 * hi];

    for (int k0 = 0; k0 < D_; k0 += 4) {
        v2f a, bm;
        // A fragment (16x4 f32): VGPR0 = K=k0+koff, VGPR1 = K=k0+koff+1
        a[0] = W[m * D_ + k0 + koff];
        a[1] = W[m * D_ + k0 + koff + 1];
        // B fragment (4x16 f32): same K striping, column n0+m
        bm[0] = xb[(size_t)(k0 + koff)     * N_ + n0 + m];
        bm[1] = xb[(size_t)(k0 + koff + 1) * N_ + n0 + m];
        // D = A*B + C   (neg_a, A, neg_b, B, c_mod, C, reuse_a, reuse_b)
        c = __builtin_amdgcn_wmma_f32_16x16x4_f32(
                false, a, false, bm, (short)0, c, false, false);
    }

    // Spill the val tile to workspace: val_ws[(b*R + r)*N + n]
    // (Re-read once by the sort kernel -> keep regular-temporal so it lands in L2.)
    float* vb = val_ws + (size_t)b * R_ * N_;
#pragma unroll
    for (int v = 0; v < 8; ++v) {
        const int r = v + 8 * hi;
        vb[(size_t)r * N_ + n0 + m] = c[v];
    }

    // argmax over the 8 local rows, then merge halves via shfl_xor(16).
    float best  = c[0];
    int   bestr = 8 * hi;
#pragma unroll
    for (int v = 1; v < 8; ++v) {
        const int r = v + 8 * hi;
        if (c[v] > best) { best = c[v]; bestr = r; }
    }
    const float obest  = __shfl_xor(best, 16, 32);
    const int   obestr = __shfl_xor(bestr, 16, 32);
    if (obest > best || (obest == best && obestr < bestr)) {
        best = obest; bestr = obestr;
    }
    if (lane < 16)
        idx_sort_out[(size_t)b * N_ + n0 + m] = (float)bestr;
}

// ---------------------------------------------------------------------------
// Kernel 2: per-(b,r) stable-descending bitonic sort of 8192 (val, idx) pairs
// in LDS (64 KB). Emits top-k indices to workspace (int) and the broadcast
// idx_star output (float, non-temporal: 32 MB write-once stream).
// One workgroup per row; 256 workgroups.
// Comparator: ascending in key (-val, idx)  ==  stable descending by value.
// ---------------------------------------------------------------------------
__global__ __launch_bounds__(1024) void softpool_topk_sort(
    const float* __restrict__ val_ws, int* __restrict__ idx_top_ws,
    float* __restrict__ idx_star_out)
{
    __shared__ float sv[N_];
    __shared__ int   si[N_];

    const int br = blockIdx.x;          // b*R + r
    const int b  = br / R_;
    const int r  = br % R_;

    const float* row = val_ws + (size_t)br * N_;
    for (int i = threadIdx.x; i < N_; i += blockDim.x) { sv[i] = row[i]; si[i] = i; }
    __syncthreads();

    for (int ksz = 2; ksz <= N_; ksz <<= 1) {
        for (int j = ksz >> 1; j > 0; j >>= 1) {
            for (int l = threadIdx.x; l < N_; l += blockDim.x) {
                const int p = l ^ j;
                if (p > l) {
                    const bool  asc = ((l & ksz) == 0);
                    const float va = sv[l], vp = sv[p];
                    const int   ia = si[l], ip = si[p];
                    // less(a,b) in (-val, idx) order; keys unique (idx unique)
                    const bool a_less_p = (va > vp) || (va == vp && ia < ip);
                    if (asc ? !a_less_p : a_less_p) {
                        sv[l] = vp; sv[p] = va;
                        si[l] = ip; si[p] = ia;
                    }
                }
            }
            __syncthreads();
        }
    }

    // First k entries are the stable-descending top-k.
    for (int jj = threadIdx.x; jj < K_; jj += blockDim.x) {
        const int idx = si[jj];
        idx_top_ws[(size_t)br * K_ + jj] = idx;   // small, re-read by gather: RT
        const float fidx = (float)idx;
#pragma unroll
        for (int i2 = 0; i2 < R_; ++i2)
            __builtin_nontemporal_store(
                fidx, &idx_star_out[(((size_t)b * R_ + i2) * R_ + r) * K_ + jj]);
    }
}

// ---------------------------------------------------------------------------
// Kernel 3: FEAT_star[b,d,r,j] = x[b,d,idx_top[b,r,j]]
// 512 MB of b128 stores (bandwidth-dominant stage). Stores are non-temporal
// so the write-once stream does not evict x (128 MB, L2-resident) which the
// random per-row gathers depend on. b128 index loads stay coalesced.
// ---------------------------------------------------------------------------
__global__ __launch_bounds__(256) void softpool_gather(
    const float* __restrict__ x, const int* __restrict__ idx_top_ws,
    float* __restrict__ feat_out)
{
    const size_t tid = (size_t)blockIdx.x * blockDim.x + threadIdx.x;
    // tid = ((b*D + d)*R + r)*(K/4) + t
    const int t    = (int)(tid & (K_ / 4 - 1));
    size_t rest    = tid / (K_ / 4);
    const int r    = (int)(rest & (R_ - 1));
    rest >>= 4;
    const int d    = (int)(rest & (D_ - 1));
    const int b    = (int)(rest >> 8);

    const v4i* ip  = (const v4i*)(idx_top_ws + (size_t)(b * R_ + r) * K_);
    const v4i  id4 = ip[t];

    const float* xr = x + ((size_t)b * D_ + d) * N_;
    v4f o;
    o.x = xr[id4.x];
    o.y = xr[id4.y];
    o.z = xr[id4.z];
    o.w = xr[id4.w];
    __builtin_nontemporal_store(o, &((v4f*)feat_out)[tid]);
}

// ---------------------------------------------------------------------------
extern "C" void kernel_launch(void* const* d_in, const int* in_sizes, int n_in,
                              void* d_out, int out_size, void* d_ws, size_t ws_size,
                              hipStream_t stream)
{
    (void)in_sizes; (void)n_in; (void)out_size; (void)ws_size;

    const float* x    = (const float*)d_in[0];   // [B, D, N]
    const float* W    = (const float*)d_in[1];   // [R, D]
    const float* bias = (const float*)d_in[2];   // [R]

    float* out          = (float*)d_out;
    float* feat_out     = out;                                        // [B,D,R,K]
    float* idx_star_out = out + (size_t)B_ * D_ * R_ * K_;            // [B,R,R,K]
    float* idx_sort_out = idx_star_out + (size_t)B_ * R_ * R_ * K_;   // [B,N]

    float* val_ws     = (float*)d_ws;                                  // [B,R,N] f32 (8 MB)
    int*   idx_top_ws = (int*)((char*)d_ws +
                               (size_t)B_ * R_ * N_ * sizeof(float));  // [B,R,K] i32 (2 MB)

    // Stage 1: WMMA GEMM + argmax
    softpool_gemm_argmax<<<B_ * (N_ / 16), 32, 0, stream>>>(
        x, W, bias, val_ws, idx_sort_out);

    // Stage 2: per-row bitonic top-k (sorted, stable-descending)
    softpool_topk_sort<<<B_ * R_, 1024, 0, stream>>>(
        val_ws, idx_top_ws, idx_star_out);

    // Stage 3: bandwidth-bound gather into FEAT_star (NT stores)
    const size_t threads3 = (size_t)B_ * D_ * R_ * (K_ / 4);
    softpool_gather<<<(unsigned)(threads3 / 256), 256, 0, stream>>>(
        x, idx_top_ws, feat_out);
}